// dRMSD_32787780338177
// MI455X (gfx1250) — compile-verified
//
#include <hip/hip_runtime.h>

typedef float v2f __attribute__((ext_vector_type(2)));
typedef float v8f __attribute__((ext_vector_type(8)));

#define LRES  1024
#define BATCH 64
#define NT    (LRES / 16)            // 64 tiles per dimension
#define NPAIR (NT * (NT + 1) / 2)    // 2080 upper-triangular tile pairs

// relu + sqrt in two raw instructions: v_med3_num_f32 + v_sqrt_f32
__device__ __forceinline__ float relu_sqrt(float x) {
    return __builtin_amdgcn_sqrtf(
        __builtin_amdgcn_fmed3f(x, 0.0f, __FLT_MAX__));
}

// ---------------------------------------------------------------------------
// Kernel 1: repack [L,B,3] -> [B,L,4] float4 (w = |p|^2) + y-validity mask.
// ---------------------------------------------------------------------------
__global__ void drmsd_repack(const float* __restrict__ x,
                             const float* __restrict__ y,
                             float4* __restrict__ xp,
                             float4* __restrict__ yp,
                             float* __restrict__ maskf) {
    int idx = blockIdx.x * blockDim.x + threadIdx.x;   // idx = l*B + b
    if (idx >= LRES * BATCH) return;
    int l = idx / BATCH;
    int b = idx - l * BATCH;
    const float* xs = x + (size_t)idx * 3;
    const float* ys = y + (size_t)idx * 3;
    float xv = xs[0], xw = xs[1], xz = xs[2];
    float yv = ys[0], yw = ys[1], yz = ys[2];
    xp[(size_t)b * LRES + l] = make_float4(xv, xw, xz, xv * xv + xw * xw + xz * xz);
    yp[(size_t)b * LRES + l] = make_float4(yv, yw, yz, yv * yv + yw * yw + yz * yz);
    maskf[(size_t)b * LRES + l] = ((yv + yw + yz) != 0.0f) ? 1.0f : 0.0f;
}

// ---------------------------------------------------------------------------
// Two chained V_WMMA_F32_16X16X4_F32 produce the full squared-distance tile:
//   WMMA1:  A=(x,y,z,sq_m)  B=(-2x,-2y,-2z,1)  ->  sq_m - 2*dot
//   WMMA2:  A=(1,0,0,0)     B=(sq_n,0,0,0)     ->  + sq_n
// A 16x4 f32 layout: V0 = K0|K2 (lanes 0-15|16-31), V1 = K1|K3.
// B 4x16 f32 layout: V0 = K0 row|K2 row,           V1 = K1 row|K3 row.
// ---------------------------------------------------------------------------
__device__ __forceinline__ v8f gram_d2(float4 pa, float4 pb, bool hi) {
    v2f a1, b1, a2, b2;
    a1[0] = hi ? pa.z : pa.x;
    a1[1] = hi ? pa.w : pa.y;
    b1[0] = -2.0f * (hi ? pb.z : pb.x);
    b1[1] = hi ? 1.0f : (-2.0f * pb.y);
    a2[0] = hi ? 0.0f : 1.0f;
    a2[1] = 0.0f;
    b2[0] = hi ? 0.0f : pb.w;
    b2[1] = 0.0f;
    v8f c = {};
    c = __builtin_amdgcn_wmma_f32_16x16x4_f32(false, a1, false, b1,
                                              (short)0, c, false, false);
    c = __builtin_amdgcn_wmma_f32_16x16x4_f32(false, a2, false, b2,
                                              (short)0, c, false, false);
    return c;   // c[v] = d2[mbase+v][lane&15]
}

// ---------------------------------------------------------------------------
// Kernel 2: one wave32 per (batch, 16x16 tile-pair), upper triangle only.
// ---------------------------------------------------------------------------
__global__ void drmsd_tiles(const float4* __restrict__ xp,
                            const float4* __restrict__ yp,
                            const float* __restrict__ maskf,
                            float* __restrict__ batch_acc) {
    const int wave = threadIdx.x >> 5;
    const int lane = threadIdx.x & 31;
    const int b    = blockIdx.y;
    const int t    = blockIdx.x * (blockDim.x >> 5) + wave;   // tile-pair id

    __shared__ float red[8];
    float acc = 0.0f;

    if (t < NPAIR) {                      // wave-uniform (always true: 260*8 == 2080)
        // Decode triangular index t -> (ti, tj), ti <= tj.
        int ti = (int)((129.0f - sqrtf(129.0f * 129.0f - 8.0f * (float)t)) * 0.5f);
        while ((ti + 1) * NT - ((ti + 1) * ti) / 2 <= t) ti++;
        while (ti * NT - (ti * (ti - 1)) / 2 > t) ti--;
        const int tj = ti + (t - (ti * NT - (ti * (ti - 1)) / 2));
        const int i0 = ti * 16, j0 = tj * 16;

        const int  r  = lane & 15;
        const bool hi = lane >= 16;
        const int  mbase = hi ? 8 : 0;

        const float4* xb = xp + (size_t)b * LRES;
        const float4* yb = yp + (size_t)b * LRES;
        const float4 pax = xb[i0 + r];   // row points (A side), .w = |p|^2
        const float4 pbx = xb[j0 + r];   // col points (B side)
        const float4 pay = yb[i0 + r];
        const float4 pby = yb[j0 + r];

        const v8f d2x = gram_d2(pax, pbx, hi);
        const v8f d2y = gram_d2(pay, pby, hi);

        // Row-side y-mask for this lane's 8 rows: 8 consecutive floats,
        // 16B aligned -> two broadcast float4 loads (no cross-lane shuffles).
        const float4* mrow =
            (const float4*)(maskf + (size_t)b * LRES + i0 + mbase);
        const float4 mlo = mrow[0];
        const float4 mhi = mrow[1];
        const float miv[8] = { mlo.x, mlo.y, mlo.z, mlo.w,
                               mhi.x, mhi.y, mhi.z, mhi.w };

        // Column-side mask * symmetry weight (off-diagonal tiles count twice).
        const float w   = (ti == tj) ? 1.0f : 2.0f;
        const float mjw = (((pby.x + pby.y + pby.z) != 0.0f) ? 1.0f : 0.0f) * w;
        const bool  diag = (ti == tj);
        const int   vd   = r - mbase;     // diagonal element's v (may be out of range)

        #pragma unroll
        for (int v = 0; v < 8; ++v) {
            const float diff = relu_sqrt(d2y[v]) - relu_sqrt(d2x[v]);
            float msk = miv[v] * mjw;
            if (diag && v == vd) msk = 0.0f;   // exclude diagonal pairs
            acc = fmaf(msk * diff, diff, acc);
        }
    }

    // Wave reduction, then block reduction, one atomic per block.
    for (int off = 16; off > 0; off >>= 1)
        acc += __shfl_down(acc, off, 32);
    if (lane == 0) red[wave] = acc;
    __syncthreads();
    if (threadIdx.x == 0) {
        float s = 0.0f;
        const int nw = blockDim.x >> 5;
        for (int wv = 0; wv < nw; ++wv) s += red[wv];
        atomicAdd(&batch_acc[b], s);
    }
}

// ---------------------------------------------------------------------------
// Kernel 3: out = sum_b sqrt(acc[b]) / sqrt(L^2/2 - L) / B
// ---------------------------------------------------------------------------
__global__ void drmsd_finalize(const float* __restrict__ batch_acc,
                               float* __restrict__ out) {
    __shared__ float red[BATCH];
    const int tid = threadIdx.x;
    red[tid] = sqrtf(batch_acc[tid]);
    __syncthreads();
    if (tid == 0) {
        float s = 0.0f;
        for (int i = 0; i < BATCH; ++i) s += red[i];
        const float denom = 723.3698936f;   // sqrt(1024*1024/2 - 1024)
        out[0] = s / denom / (float)BATCH;
    }
}

// ---------------------------------------------------------------------------
extern "C" void kernel_launch(void* const* d_in, const int* in_sizes, int n_in,
                              void* d_out, int out_size, void* d_ws, size_t ws_size,
                              hipStream_t stream) {
    const float* x = (const float*)d_in[0];
    const float* y = (const float*)d_in[1];
    float* out = (float*)d_out;

    char* ws = (char*)d_ws;
    const size_t ptsBytes  = (size_t)BATCH * LRES * sizeof(float4);  // 1 MB each
    const size_t maskBytes = (size_t)BATCH * LRES * sizeof(float);   // 256 KB
    float4* xp        = (float4*)ws;
    float4* yp        = (float4*)(ws + ptsBytes);
    float*  maskf     = (float*)(ws + 2 * ptsBytes);
    float*  batch_acc = (float*)(ws + 2 * ptsBytes + maskBytes);

    hipMemsetAsync(batch_acc, 0, BATCH * sizeof(float), stream);

    drmsd_repack<<<(LRES * BATCH + 255) / 256, 256, 0, stream>>>(x, y, xp, yp, maskf);

    dim3 grid((NPAIR + 7) / 8, BATCH);   // 260 x 64 blocks, 8 waves/block
    drmsd_tiles<<<grid, 256, 0, stream>>>(xp, yp, maskf, batch_acc);

    drmsd_finalize<<<1, BATCH, 0, stream>>>(batch_acc, out);
}